// HGTGNN_47828755808357
// MI455X (gfx1250) — compile-verified
//
#include <hip/hip_runtime.h>
#include <math.h>

// ---- problem constants (match reference) ----
#define NN   50000     // nodes  (multiple of 16 -> no row guards in GEMM)
#define NE   500000    // edges
#define TT   3         // node types
#define RR   4         // relation types
#define HH   8         // heads
#define DNH  128       // model dim
#define LL   2         // layers
#define DINF 166       // input feature dim
#define DINP 168       // padded to multiple of 4
#define MAXTT 240      // time positions
#define DKK  16        // head dim

typedef __attribute__((ext_vector_type(2))) float v2f;
typedef __attribute__((ext_vector_type(8))) float v8f;

// =====================================================================
// Per-type linear:  Y[n,0:128] = epilogue( X[n,0:KD] @ W[type(n)] + b[type(n)] )
// WMMA f32 16x16x4.  Block = 128 threads (4 waves) owns one 16-row node
// tile; each wave computes two 16x16 output col-tiles (A-frag reuse).
// A-tile staged in LDS zero-padded to KDP => branch-free inner loop.
// W must already be padded to [T, KDP, 128] (zero rows KD..KDP).
// mode: 0 = none, 1 = tanh, 2 = blend: y = v*sig(skip[t]) + Hold*(1-sig)
// =====================================================================
template<int KD, int KDP>
__global__ __launch_bounds__(128)
void ptl_wmma(const float* __restrict__ X, const int* __restrict__ types,
              const float* __restrict__ W,   // [T, KDP, 128], padded
              const float* __restrict__ B,   // [T, 128]
              float* __restrict__ Y, int mode,
              const float* __restrict__ Hold, const float* __restrict__ skipv)
{
    __shared__ float Xs[16][KDP];
    __shared__ int   ts[16];
    const int tid  = threadIdx.x;
    const int row0 = blockIdx.x * 16;

    for (int idx = tid; idx < 16 * KDP; idx += 128) {
        const int r = idx / KDP, k = idx - r * KDP;
        Xs[r][k] = (k < KD) ? X[(size_t)(row0 + r) * KD + k] : 0.0f;
    }
    if (tid < 16) ts[tid] = types[row0 + tid];
    __syncthreads();

    const int lane = tid & 31;
    const int wave = tid >> 5;
    const int half = lane >> 4;       // 0: K=k,k+1   1: K=k+2,k+3
    const int l15  = lane & 15;
    const int col0 = wave * 32 + l15; // this wave's first col tile
    const int col1 = col0 + 16;       // second col tile

    for (int t = 0; t < TT; ++t) {
        const float* Wt = W + (size_t)t * KDP * DNH;
        v8f acc0 = {}, acc1 = {};
        #pragma unroll 4
        for (int k = 0; k < KDP; k += 4) {
            const int ka = k + half * 2;
            const v2f a = *(const v2f*)&Xs[l15][ka];   // ds_load_b64
            v2f b0, b1;
            b0.x = Wt[(size_t)ka       * DNH + col0];
            b0.y = Wt[(size_t)(ka + 1) * DNH + col0];
            b1.x = Wt[(size_t)ka       * DNH + col1];
            b1.y = Wt[(size_t)(ka + 1) * DNH + col1];
            acc0 = __builtin_amdgcn_wmma_f32_16x16x4_f32(
                       false, a, false, b0, (short)0, acc0, false, false);
            acc1 = __builtin_amdgcn_wmma_f32_16x16x4_f32(
                       false, a, false, b1, (short)0, acc1, false, false);
        }
        const int   rbase  = half * 8;
        const float bias0  = B[t * DNH + col0];
        const float bias1  = B[t * DNH + col1];
        float alpha = 0.0f;
        if (mode == 2) alpha = 1.0f / (1.0f + __expf(-skipv[t]));
        #pragma unroll
        for (int i = 0; i < 8; ++i) {
            if (ts[rbase + i] != t) continue;
            const size_t row = row0 + rbase + i;
            float v0 = acc0[i] + bias0;
            float v1 = acc1[i] + bias1;
            if (mode == 1) { v0 = tanhf(v0); v1 = tanhf(v1); }
            else if (mode == 2) {
                v0 = v0 * alpha + Hold[row * DNH + col0] * (1.0f - alpha);
                v1 = v1 * alpha + Hold[row * DNH + col1] * (1.0f - alpha);
            }
            Y[row * DNH + col0] = v0;
            Y[row * DNH + col1] = v1;
        }
    }
}

// zero-pad adapter weights [T,166,128] -> [T,168,128]
__global__ __launch_bounds__(256)
void pad_weights(const float* __restrict__ W, float* __restrict__ Wp)
{
    const int idx = blockIdx.x * 256 + threadIdx.x;
    if (idx >= TT * DINP * DNH) return;
    const int j = idx % DNH;
    const int k = (idx / DNH) % DINP;
    const int t = idx / (DNH * DINP);
    Wp[idx] = (k < DINF) ? W[((size_t)t * DINF + k) * DNH + j] : 0.0f;
}

// =====================================================================
// rte_proj[p,0:128] = sinusoid_tab[p] @ rteW + rteb   (240 x 128, tiny)
// =====================================================================
__global__ __launch_bounds__(DNH)
void build_rte_proj(const float* __restrict__ rteW, const float* __restrict__ rteb,
                    float* __restrict__ out)
{
    __shared__ float tab[DNH];
    const int p = blockIdx.x, j = threadIdx.x;
    const int i = j >> 1;
    const float div = __expf(-(float)(2 * i) * (logf(10000.0f) / (float)DNH));
    const float ang = (float)p * div;
    tab[j] = (((j & 1) == 0) ? sinf(ang) : cosf(ang)) * 0.08838834764831845f;
    __syncthreads();
    float s = rteb[j];
    for (int k = 0; k < DNH; ++k) s += tab[k] * rteW[k * DNH + j];
    out[p * DNH + j] = s;
}

// ktab[t][p] = rte_proj[p] @ kW_l[t] ; vtab analogous (z dim picks which)
__global__ __launch_bounds__(DNH)
void time_tables(const float* __restrict__ rte_proj,
                 const float* __restrict__ kW_l, const float* __restrict__ vW_l,
                 float* __restrict__ ktab, float* __restrict__ vtab)
{
    const int p = blockIdx.x, t = blockIdx.y, which = blockIdx.z, j = threadIdx.x;
    const float* W = (which == 0 ? kW_l : vW_l) + (size_t)t * DNH * DNH;
    float* out     = (which == 0 ? ktab : vtab);
    __shared__ float rp[DNH];
    rp[j] = rte_proj[p * DNH + j];
    __syncthreads();
    float s = 0.0f;
    for (int k = 0; k < DNH; ++k) s += rp[k] * W[k * DNH + j];
    out[((size_t)t * MAXTT + p) * DNH + j] = s;
}

__device__ __forceinline__ unsigned fmap(float x) {
    unsigned u = __float_as_uint(x);
    return (u & 0x80000000u) ? ~u : (u | 0x80000000u);
}
__device__ __forceinline__ float funmap(unsigned u) {
    return __uint_as_float((u & 0x80000000u) ? (u ^ 0x80000000u) : ~u);
}
__device__ __forceinline__ long long fx32(float x) {   // Q32.32 fixed point
    return llrintf(x * 4294967296.0f);
}

// =====================================================================
// Edge pass A: att[e,h] = (q[tgt,h] . (k_e @ rel_att[r,h])) * pri[r,h]/4
// + order-invariant atomic segment-max.  8 threads per edge (1 per head);
// all 16-float streams read as float4 (global_load_b128, 64B aligned).
// =====================================================================
__global__ __launch_bounds__(256)
void edge_att(const float* __restrict__ Q, const float* __restrict__ K0,
              const float* __restrict__ ktab,
              const int* __restrict__ src, const int* __restrict__ tgt,
              const int* __restrict__ etype, const int* __restrict__ etime,
              const int* __restrict__ ntype,
              const float* __restrict__ relA, const float* __restrict__ pri,
              float* __restrict__ att, unsigned* __restrict__ nmax, int E)
{
    __shared__ float sA[RR * HH * DKK * DKK];
    __shared__ float sP[RR * HH];
    for (int i = threadIdx.x; i < RR * HH * DKK * DKK; i += 256) sA[i] = relA[i];
    for (int i = threadIdx.x; i < RR * HH; i += 256) sP[i] = pri[i];
    __syncthreads();

    const int e = blockIdx.x * 32 + (threadIdx.x >> 3);
    const int h = threadIdx.x & 7;
    if (e >= E) return;
    const int s = src[e], g = tgt[e], r = etype[e], tm = etime[e];
    const int st = ntype[s];

    const float4* k04 = (const float4*)(K0   + (size_t)s * DNH + h * DKK);
    const float4* kt4 = (const float4*)(ktab + ((size_t)st * MAXTT + tm) * DNH + h * DKK);
    const float4* qv4 = (const float4*)(Q    + (size_t)g * DNH + h * DKK);
    float kv[DKK], q[DKK];
    #pragma unroll
    for (int c = 0; c < 4; ++c) {
        const float4 a = k04[c], b = kt4[c], qq = qv4[c];
        kv[4*c+0] = a.x + b.x; kv[4*c+1] = a.y + b.y;
        kv[4*c+2] = a.z + b.z; kv[4*c+3] = a.w + b.w;
        q[4*c+0] = qq.x; q[4*c+1] = qq.y; q[4*c+2] = qq.z; q[4*c+3] = qq.w;
    }

    const float* A = sA + (r * HH + h) * DKK * DKK;
    float a = 0.0f;
    #pragma unroll
    for (int f = 0; f < DKK; ++f) {
        float kr = 0.0f;
        #pragma unroll
        for (int d = 0; d < DKK; ++d) kr += kv[d] * A[d * DKK + f];
        a += q[f] * kr;
    }
    a *= sP[r * HH + h] * 0.25f;                 // / sqrt(16)
    att[(size_t)e * HH + h] = a;
    atomicMax(nmax + (size_t)g * HH + h, fmap(a));
}

// =====================================================================
// Edge pass B: ex = exp(att - max); den[g,h] += ex ;
// num[g,h*16+f] += ex * (v_e @ rel_msg[r,h])  in Q32.32 integer fixed
// point (associative atomics -> bitwise deterministic).
// =====================================================================
__global__ __launch_bounds__(256)
void edge_agg(const float* __restrict__ V0, const float* __restrict__ vtab,
              const float* __restrict__ att, const unsigned* __restrict__ nmax,
              const int* __restrict__ src, const int* __restrict__ tgt,
              const int* __restrict__ etype, const int* __restrict__ etime,
              const int* __restrict__ ntype,
              const float* __restrict__ relM,
              unsigned long long* __restrict__ num,
              unsigned long long* __restrict__ den, int E)
{
    __shared__ float sM[RR * HH * DKK * DKK];
    for (int i = threadIdx.x; i < RR * HH * DKK * DKK; i += 256) sM[i] = relM[i];
    __syncthreads();

    const int e = blockIdx.x * 32 + (threadIdx.x >> 3);
    const int h = threadIdx.x & 7;
    if (e >= E) return;
    const int s = src[e], g = tgt[e], r = etype[e], tm = etime[e];
    const int st = ntype[s];

    const float m  = funmap(nmax[(size_t)g * HH + h]);
    const float ex = __expf(att[(size_t)e * HH + h] - m);
    atomicAdd(den + (size_t)g * HH + h, (unsigned long long)fx32(ex));

    const float4* v04 = (const float4*)(V0   + (size_t)s * DNH + h * DKK);
    const float4* vt4 = (const float4*)(vtab + ((size_t)st * MAXTT + tm) * DNH + h * DKK);
    float vv[DKK];
    #pragma unroll
    for (int c = 0; c < 4; ++c) {
        const float4 a = v04[c], b = vt4[c];
        vv[4*c+0] = a.x + b.x; vv[4*c+1] = a.y + b.y;
        vv[4*c+2] = a.z + b.z; vv[4*c+3] = a.w + b.w;
    }

    const float* M = sM + (r * HH + h) * DKK * DKK;
    #pragma unroll
    for (int f = 0; f < DKK; ++f) {
        float vr = 0.0f;
        #pragma unroll
        for (int d = 0; d < DKK; ++d) vr += vv[d] * M[d * DKK + f];
        atomicAdd(num + (size_t)g * DNH + h * DKK + f,
                  (unsigned long long)fx32(ex * vr));
    }
}

// Node pass: agg = num/max(den,1e-9) ; G = gelu_exact(agg)
__global__ __launch_bounds__(256)
void node_gelu(const unsigned long long* __restrict__ num,
               const unsigned long long* __restrict__ den,
               float* __restrict__ G, int total)
{
    const int idx = blockIdx.x * 256 + threadIdx.x;
    if (idx >= total) return;
    const int nrow = idx >> 7, j = idx & 127, h = j >> 4;
    const float inv32 = 2.3283064365386963e-10f;    // 2^-32
    float dn = (float)(long long)den[(size_t)nrow * HH + h] * inv32;
    dn = fmaxf(dn, 1e-9f);
    const float x = ((float)(long long)num[idx] * inv32) / dn;
    G[idx] = 0.5f * x * (1.0f + erff(x * 0.7071067811865475f));
}

// =====================================================================
extern "C" void kernel_launch(void* const* d_in, const int* in_sizes, int n_in,
                              void* d_out, int out_size, void* d_ws, size_t ws_size,
                              hipStream_t stream)
{
    const float* node_feature = (const float*)d_in[0];
    const int*   node_type    = (const int*)  d_in[1];
    const int*   edge_time    = (const int*)  d_in[2];
    const int*   edge_index   = (const int*)  d_in[3];
    const int*   edge_type    = (const int*)  d_in[4];
    const float* adapt_W = (const float*)d_in[5];
    const float* adapt_b = (const float*)d_in[6];
    const float* kW  = (const float*)d_in[7];
    const float* kb  = (const float*)d_in[8];
    const float* qW  = (const float*)d_in[9];
    const float* qb  = (const float*)d_in[10];
    const float* vW  = (const float*)d_in[11];
    const float* vb  = (const float*)d_in[12];
    const float* aW  = (const float*)d_in[13];
    const float* ab  = (const float*)d_in[14];
    const float* rel_att = (const float*)d_in[15];
    const float* rel_msg = (const float*)d_in[16];
    const float* rel_pri = (const float*)d_in[17];
    const float* skip    = (const float*)d_in[18];
    const float* rteW    = (const float*)d_in[19];
    const float* rteb    = (const float*)d_in[20];

    const int* src = edge_index;           // edge_index[0]
    const int* tgt = edge_index + NE;      // edge_index[1]

    // ---- carve scratch out of d_ws ----
    char* ws = (char*)d_ws;
    size_t off = 0;
    auto carve = [&](size_t bytes) -> void* {
        void* p = ws + off;
        off = (off + bytes + 255) & ~(size_t)255;
        return p;
    };
    float* HA       = (float*)carve((size_t)NN * DNH * 4);
    float* HB       = (float*)carve((size_t)NN * DNH * 4);
    float* Q        = (float*)carve((size_t)NN * DNH * 4);   // reused as G
    float* K0       = (float*)carve((size_t)NN * DNH * 4);
    float* V0       = (float*)carve((size_t)NN * DNH * 4);
    float* Wpad     = (float*)carve((size_t)TT * DINP * DNH * 4);
    float* rte_proj = (float*)carve((size_t)MAXTT * DNH * 4);
    float* ktab     = (float*)carve((size_t)TT * MAXTT * DNH * 4);
    float* vtab     = (float*)carve((size_t)TT * MAXTT * DNH * 4);
    float* att      = (float*)carve((size_t)NE * HH * 4);
    unsigned* nmax  = (unsigned*)carve((size_t)NN * HH * 4);
    unsigned long long* den = (unsigned long long*)carve((size_t)NN * HH * 8);
    unsigned long long* num = (unsigned long long*)carve((size_t)NN * DNH * 8);
    float* G = Q;   // Q is dead after edge pass A; reuse for gelu output

    const int gGemm = NN / 16;             // 3125 node tiles, block 128
    const int gEdge = (NE + 31) / 32;      // 32 edges per 256-thread block

    // ---- adapter: h0 = tanh(per_type_linear(x)) ----
    pad_weights<<<(TT * DINP * DNH + 255) / 256, 256, 0, stream>>>(adapt_W, Wpad);
    ptl_wmma<DINF, DINP><<<gGemm, 128, 0, stream>>>(node_feature, node_type,
                                                    Wpad, adapt_b, HA, 1,
                                                    nullptr, nullptr);

    for (int l = 0; l < LL; ++l) {
        const float* kW_l  = kW  + (size_t)l * TT * DNH * DNH;
        const float* kb_l  = kb  + (size_t)l * TT * DNH;
        const float* qW_l  = qW  + (size_t)l * TT * DNH * DNH;
        const float* qb_l  = qb  + (size_t)l * TT * DNH;
        const float* vW_l  = vW  + (size_t)l * TT * DNH * DNH;
        const float* vb_l  = vb  + (size_t)l * TT * DNH;
        const float* aW_l  = aW  + (size_t)l * TT * DNH * DNH;
        const float* ab_l  = ab  + (size_t)l * TT * DNH;
        const float* rA_l  = rel_att + (size_t)l * RR * HH * DKK * DKK;
        const float* rM_l  = rel_msg + (size_t)l * RR * HH * DKK * DKK;
        const float* pri_l = rel_pri + (size_t)l * RR * HH;
        const float* skp_l = skip + (size_t)l * TT;
        const float* rW_l  = rteW + (size_t)l * DNH * DNH;
        const float* rb_l  = rteb + (size_t)l * DNH;

        float* Hin  = (l == 0) ? HA : HB;
        float* Hout = (l == 0) ? HB : (float*)d_out;

        build_rte_proj<<<MAXTT, DNH, 0, stream>>>(rW_l, rb_l, rte_proj);

        ptl_wmma<DNH, DNH><<<gGemm, 128, 0, stream>>>(Hin, node_type, qW_l, qb_l, Q,  0, nullptr, nullptr);
        ptl_wmma<DNH, DNH><<<gGemm, 128, 0, stream>>>(Hin, node_type, kW_l, kb_l, K0, 0, nullptr, nullptr);
        ptl_wmma<DNH, DNH><<<gGemm, 128, 0, stream>>>(Hin, node_type, vW_l, vb_l, V0, 0, nullptr, nullptr);

        time_tables<<<dim3(MAXTT, TT, 2), DNH, 0, stream>>>(rte_proj, kW_l, vW_l, ktab, vtab);

        hipMemsetAsync(nmax, 0, (size_t)NN * HH * 4, stream);
        hipMemsetAsync(den,  0, (size_t)NN * HH * 8, stream);
        hipMemsetAsync(num,  0, (size_t)NN * DNH * 8, stream);

        edge_att<<<gEdge, 256, 0, stream>>>(Q, K0, ktab, src, tgt, edge_type,
                                            edge_time, node_type, rA_l, pri_l,
                                            att, nmax, NE);
        edge_agg<<<gEdge, 256, 0, stream>>>(V0, vtab, att, nmax, src, tgt,
                                            edge_type, edge_time, node_type,
                                            rM_l, num, den, NE);
        node_gelu<<<(NN * DNH + 255) / 256, 256, 0, stream>>>(num, den, G, NN * DNH);

        ptl_wmma<DNH, DNH><<<gGemm, 128, 0, stream>>>(G, node_type, aW_l, ab_l, Hout,
                                                      2, Hin, skp_l);
    }
}